// quadLayer_63342177681649
// MI455X (gfx1250) — compile-verified
//
#include <hip/hip_runtime.h>

typedef float v2f __attribute__((ext_vector_type(2)));
typedef float v8f __attribute__((ext_vector_type(8)));

#define NCOLS 512          // C*FM = 16*32 output columns of the fused GEMM
#define RTOT  20           // K = 4 (linear) + 16 (quadratic)
#define LOCS_PER_BLOCK 32  // 8192 locations / 256 blocks

__global__ __launch_bounds__(256, 1) void quad_wmma_kernel(
    const float* __restrict__ x,   // [8,64,64,16]
    const float* __restrict__ W1,  // [16,4,32]
    const float* __restrict__ W2,  // [16,16,32]
    float* __restrict__ out)       // [8192, 16, 512] as (loc, c, a*32+f) -> loc*8192 + c*512 + a*32 + f
{
    // Wcat^T in LDS: ldsW[col*20 + r], col = c*32+f, r<4 -> W1[c,r,f], else W2[c,r-4,f]
    __shared__ float ldsW[NCOLS * RTOT];  // 40 KB

    const int tid = threadIdx.x;
    for (int idx = tid; idx < NCOLS * RTOT; idx += 256) {
        int col = idx / RTOT;
        int r   = idx - col * RTOT;
        int c   = col >> 5;
        int f   = col & 31;
        ldsW[idx] = (r < 4) ? W1[(c * 4 + r) * 32 + f]
                            : W2[(c * 16 + (r - 4)) * 32 + f];
    }
    __syncthreads();

    const int lane = tid & 31;
    const int wave = tid >> 5;   // 0..7
    const int n    = lane & 15;  // B/D column within 16-wide tile; also A row index
    const int hi   = lane >> 4;  // lane half selects K pair (A/B) and M+8 (D)

    // B fragments: wave covers column tiles 4*wave .. 4*wave+3, each tile needs 5 K-chunks.
    // WMMA f32 B 4x16 layout: lanes 0-15 hold K0/K1 in VGPR0/1, lanes 16-31 hold K2/K3.
    v2f Bf[4][5];
#pragma unroll
    for (int t = 0; t < 4; ++t) {
        const int ntile = wave * 4 + t;
        const int col   = ntile * 16 + n;
#pragma unroll
        for (int kc = 0; kc < 5; ++kc) {
            // byte address = (col*20 + 4*kc + 2*hi)*4, always 8B aligned -> ds_load_b64
            Bf[t][kc] = *(const v2f*)&ldsW[col * RTOT + 4 * kc + 2 * hi];
        }
    }

    const int loc0 = blockIdx.x * LOCS_PER_BLOCK;
    for (int i = 0; i < LOCS_PER_BLOCK; ++i) {
        const int loc = loc0 + i;
        const int b   = loc >> 10;
        const int rem = loc & 1023;
        const int h   = rem >> 5;
        const int w   = rem & 31;

        // Patch vector pv[64]: d=4a..4a+3 is p-row a. Rows (2h,2w..2w+1) and (2h+1,2w..2w+1)
        // are two contiguous 32-float runs. Lane loads its row's 4 floats with one b128.
        const float* row0 = x + (((b * 64 + 2 * h) * 64) + 2 * w) * 16;
        const float* row1 = row0 + 64 * 16;
        const float* src  = (n < 8) ? (row0 + 4 * n) : (row1 + 4 * n - 32);
        float4 p4 = *(const float4*)src;
        const float pk[4] = {p4.x, p4.y, p4.z, p4.w};

        // pext[a, 4kc+2hi (+1)]: kc=0 -> linear terms, kc>=1 -> p[kc-1]*p[2hi (+1)]
        const float pj0 = hi ? p4.z : p4.x;
        const float pj1 = hi ? p4.w : p4.y;
        v2f Af[5];
        Af[0].x = pj0;  Af[0].y = pj1;
#pragma unroll
        for (int kc = 1; kc < 5; ++kc) {
            Af[kc].x = pk[kc - 1] * pj0;
            Af[kc].y = pk[kc - 1] * pj1;
        }

        v8f acc[4];
#pragma unroll
        for (int t = 0; t < 4; ++t) {
            acc[t] = (v8f){0.f, 0.f, 0.f, 0.f, 0.f, 0.f, 0.f, 0.f};
#pragma unroll
            for (int kc = 0; kc < 5; ++kc) {
                acc[t] = __builtin_amdgcn_wmma_f32_16x16x4_f32(
                    false, Af[kc], false, Bf[t][kc], (short)0, acc[t],
                    false, false);
            }
        }

        // D layout: VGPR v, lanes 0-15 -> (M=v, N=lane), lanes 16-31 -> (M=v+8, N=lane-16).
        // Output element index = loc*8192 + c*512 + (v+8*hi)*32 + f, with
        // c = ntile>>1, f = (ntile&1)*16 + n.
        float* obase = out + (size_t)loc * 8192 + hi * 256 + n;
#pragma unroll
        for (int t = 0; t < 4; ++t) {
            const int ntile = wave * 4 + t;
            float* op = obase + (ntile >> 1) * 512 + (ntile & 1) * 16;
#pragma unroll
            for (int v = 0; v < 8; ++v) {
                __builtin_nontemporal_store(acc[t][v], op + v * 32);
            }
        }
    }
}

extern "C" void kernel_launch(void* const* d_in, const int* in_sizes, int n_in,
                              void* d_out, int out_size, void* d_ws, size_t ws_size,
                              hipStream_t stream) {
    (void)in_sizes; (void)n_in; (void)out_size; (void)d_ws; (void)ws_size;
    const float* x  = (const float*)d_in[0];
    const float* W1 = (const float*)d_in[1];
    const float* W2 = (const float*)d_in[2];
    float* out = (float*)d_out;
    // 8192 locations, 32 per block -> 256 blocks of 8 waves
    quad_wmma_kernel<<<dim3(8192 / LOCS_PER_BLOCK), dim3(256), 0, stream>>>(x, W1, W2, out);
}